// CrossEntropy3d_Ohem_37692632989924
// MI455X (gfx1250) — compile-verified
//
#include <hip/hip_runtime.h>
#include <stdint.h>

// ---------------------------------------------------------------------------
// OHEM 3D cross-entropy for MI455X (gfx1250, wave32).
// predict: (2, 12, 64, 128, 128) f32   target: (2, 64, 128, 128) int32
// out: scalar f32 loss.
//
// Memory-bound: ~101MB predict streamed once into LDS via CDNA5 async
// global->LDS copies; 8MB key array stays in the 192MB L2 for the 4
// radix-select passes. All reductions use fixed-shape trees so the result is
// deterministic across graph replays.
// ---------------------------------------------------------------------------

#define IGNORE_LABEL 255
#define THRESH_F 0.9f
#define MIN_KEPT 10000u

static constexpr int C    = 12;
static constexpr int DHW  = 64 * 128 * 128;   // 1,048,576 (power of two)
static constexpr int NPIX = 2 * DHW;          // 2,097,152
static constexpr int RB   = 1024;             // reduce blocks

struct Ctl {
  unsigned num_valid;
  unsigned keep_all;
  unsigned prefix;
  unsigned remaining;
  float    threshold;
};

// ---------------------------------------------------------------------------
__global__ __launch_bounds__(256) void k_init(unsigned* __restrict__ hist,
                                              Ctl* __restrict__ ctl) {
  hist[threadIdx.x] = 0u;
  if (threadIdx.x == 0) ctl->num_valid = 0u;
}

// ---------------------------------------------------------------------------
// Per-pixel stable log-softmax; emit prob key (monotone uint32) + nll.
// The block's 12x256-float logit tile is staged into LDS with CDNA5 async
// global->LDS copies (ASYNCcnt path), then consumed bank-conflict-free.
__global__ __launch_bounds__(256) void k_prob_nll(const float* __restrict__ predict,
                                                  const int* __restrict__ target,
                                                  unsigned* __restrict__ keys,
                                                  float* __restrict__ nll,
                                                  Ctl* __restrict__ ctl) {
  __shared__ float tile[C * 256];                     // 12 KB

  const int tid = threadIdx.x;
  const int blockStart = blockIdx.x * 256;            // first pixel of block
  const int j   = blockStart + tid;                   // this thread's pixel
  const int n   = blockStart >> 20;                   // blockStart / DHW (block never straddles n)
  const int p0  = blockStart & (DHW - 1);             // pixel offset within (d,h,w)
  const float* gbase = predict + (size_t)n * (size_t)C * (size_t)DHW + (size_t)p0;

  // ---- async stage: 12 rows x 1KB, each row = 64 lanes x 16B --------------
  {
    const unsigned seg     = ((unsigned)tid & 63u) * 16u;  // byte offset in row
    const unsigned rowbase = (unsigned)tid >> 6;           // 0..3
    const unsigned ldsbase = (unsigned)(uintptr_t)(&tile[0]);
#pragma unroll
    for (int r = 0; r < 3; ++r) {
      const unsigned row = (unsigned)r * 4u + rowbase;     // 0..11
      const char* g = (const char*)gbase + (size_t)row * (size_t)DHW * 4u + seg;
      const unsigned l = ldsbase + row * 1024u + seg;
      asm volatile("global_load_async_to_lds_b128 %0, %1, off"
                   :: "v"(l), "v"(g)
                   : "memory");
    }
    asm volatile("s_wait_asynccnt 0x0" ::: "memory");      // own wave's copies done
  }

  const int lab = target[j];
  const bool valid = (lab != IGNORE_LABEL);
  const int slab = valid ? lab : 0;

  __syncthreads();                                         // all waves' copies visible

  float xs[C];
  float m    = -__builtin_huge_valf();
  float xlab = 0.0f;
#pragma unroll
  for (int c = 0; c < C; ++c) {
    const float x = tile[c * 256 + tid];                   // stride-256: conflict-free
    xs[c] = x;
    m = fmaxf(m, x);
    if (c == slab) xlab = x;                               // unrolled -> cndmask chain
  }
  float s = 0.0f;
#pragma unroll
  for (int c = 0; c < C; ++c) s += __expf(xs[c] - m);

  const float lp   = (xlab - m) - __logf(s);
  const float prob = __expf(lp);

  // prob >= 0 so its IEEE bits are order-preserving as uint32.
  // invalid -> 0xFFFFFFFF sorts after every valid prob (same role as +inf).
  keys[j] = valid ? __float_as_uint(prob) : 0xFFFFFFFFu;
  nll[j]  = valid ? -lp : 0.0f;

  // Valid count: wave32 ballot + per-block LDS accumulation.
  __shared__ unsigned blk;
  if (tid == 0) blk = 0u;
  __syncthreads();
  const unsigned long long b = __ballot(valid ? 1 : 0);
  if ((tid & 31) == 0)                                     // gfx1250 is wave32-only
    atomicAdd(&blk, (unsigned)__popcll(b));
  __syncthreads();
  if (tid == 0) atomicAdd(&ctl->num_valid, blk);
}

// ---------------------------------------------------------------------------
__global__ void k_setup(Ctl* __restrict__ ctl) {
  const unsigned nv = ctl->num_valid;
  unsigned r = (nv < MIN_KEPT) ? nv : MIN_KEPT;       // 1-based rank of kth value
  if (r == 0u) r = 1u;                                // k_idx = max(...-1, 0)
  ctl->remaining = r;
  ctl->prefix    = 0u;
  ctl->keep_all  = (MIN_KEPT >= nv) ? 1u : 0u;
  ctl->threshold = THRESH_F;
}

// ---------------------------------------------------------------------------
// One MSB-first radix pass: histogram of 8-bit digit among keys matching the
// already-decided prefix. Keys (8MB) are L2-resident on MI455X (192MB L2).
__global__ __launch_bounds__(256) void k_hist(const uint4* __restrict__ keys4,
                                              const Ctl* __restrict__ ctl,
                                              unsigned* __restrict__ hist,
                                              int shift) {
  __shared__ unsigned lh[256];
  lh[threadIdx.x] = 0u;
  __syncthreads();

  const unsigned prefix = ctl->prefix;
  const unsigned mask   = (shift == 24) ? 0u : (0xFFFFFFFFu << (shift + 8));

  const int i = blockIdx.x * 256 + threadIdx.x;       // NPIX/4 total elements
  const uint4 k = keys4[i];
  if ((k.x & mask) == prefix) atomicAdd(&lh[(k.x >> shift) & 255u], 1u);
  if ((k.y & mask) == prefix) atomicAdd(&lh[(k.y >> shift) & 255u], 1u);
  if ((k.z & mask) == prefix) atomicAdd(&lh[(k.z >> shift) & 255u], 1u);
  if ((k.w & mask) == prefix) atomicAdd(&lh[(k.w >> shift) & 255u], 1u);

  __syncthreads();
  const unsigned h = lh[threadIdx.x];
  if (h) atomicAdd(&hist[threadIdx.x], h);
}

// Pick the digit bucket containing the remaining rank; zero hist for next pass.
__global__ __launch_bounds__(256) void k_scan(Ctl* __restrict__ ctl,
                                              unsigned* __restrict__ hist,
                                              int shift) {
  if (threadIdx.x == 0) {
    const unsigned rem = ctl->remaining;
    unsigned cum = 0u, b = 255u;
    for (int i = 0; i < 256; ++i) {
      const unsigned h = hist[i];
      if (cum + h >= rem) { b = (unsigned)i; break; }
      cum += h;
    }
    ctl->prefix   |= b << shift;
    ctl->remaining = rem - cum;
    if (shift == 0) {
      // exact kth-smallest prob; fmaxf(NaN, 0.9f) -> 0.9f (only hit when
      // num_valid==0, where keep_all==1 makes the threshold irrelevant).
      ctl->threshold = fmaxf(__uint_as_float(ctl->prefix), THRESH_F);
    }
  }
  __syncthreads();
  hist[threadIdx.x] = 0u;
}

// ---------------------------------------------------------------------------
// Masked sum/count of nll with deterministic fixed-shape block reduction.
__global__ __launch_bounds__(256) void k_reduce(const uint4* __restrict__ keys4,
                                                const float4* __restrict__ nll4,
                                                const Ctl* __restrict__ ctl,
                                                double* __restrict__ psum,
                                                unsigned* __restrict__ pcnt) {
  const unsigned keepall = ctl->keep_all;
  const float th = ctl->threshold;

  double s = 0.0;
  unsigned cnt = 0u;
  const int total4 = NPIX / 4;
  for (int i = blockIdx.x * 256 + threadIdx.x; i < total4; i += RB * 256) {
    __builtin_prefetch(&keys4[i + RB * 256], 0, 0);   // gfx1250 global_prefetch_b8
    __builtin_prefetch(&nll4[i + RB * 256], 0, 0);
    const uint4  k = keys4[i];
    const float4 v = nll4[i];
    // valid keys are prob bits in [0, 0x3F800000]; invalid is 0xFFFFFFFF.
    if (k.x <= 0x3F800000u && (keepall || __uint_as_float(k.x) <= th)) { s += (double)v.x; ++cnt; }
    if (k.y <= 0x3F800000u && (keepall || __uint_as_float(k.y) <= th)) { s += (double)v.y; ++cnt; }
    if (k.z <= 0x3F800000u && (keepall || __uint_as_float(k.z) <= th)) { s += (double)v.z; ++cnt; }
    if (k.w <= 0x3F800000u && (keepall || __uint_as_float(k.w) <= th)) { s += (double)v.w; ++cnt; }
  }

  __shared__ double   sd[256];
  __shared__ unsigned sc[256];
  sd[threadIdx.x] = s;
  sc[threadIdx.x] = cnt;
  __syncthreads();
#pragma unroll
  for (int off = 128; off > 0; off >>= 1) {
    if (threadIdx.x < (unsigned)off) {
      sd[threadIdx.x] += sd[threadIdx.x + off];
      sc[threadIdx.x] += sc[threadIdx.x + off];
    }
    __syncthreads();
  }
  if (threadIdx.x == 0) {
    psum[blockIdx.x] = sd[0];
    pcnt[blockIdx.x] = sc[0];
  }
}

__global__ __launch_bounds__(256) void k_final(const double* __restrict__ psum,
                                               const unsigned* __restrict__ pcnt,
                                               float* __restrict__ out) {
  __shared__ double   sd[256];
  __shared__ unsigned sc[256];
  double s = 0.0;
  unsigned c = 0u;
  for (int i = threadIdx.x; i < RB; i += 256) { s += psum[i]; c += pcnt[i]; }
  sd[threadIdx.x] = s;
  sc[threadIdx.x] = c;
  __syncthreads();
#pragma unroll
  for (int off = 128; off > 0; off >>= 1) {
    if (threadIdx.x < (unsigned)off) {
      sd[threadIdx.x] += sd[threadIdx.x + off];
      sc[threadIdx.x] += sc[threadIdx.x + off];
    }
    __syncthreads();
  }
  if (threadIdx.x == 0) {
    const unsigned cc = (sc[0] == 0u) ? 1u : sc[0];
    out[0] = (float)(sd[0] / (double)cc);
  }
}

// ---------------------------------------------------------------------------
extern "C" void kernel_launch(void* const* d_in, const int* in_sizes, int n_in,
                              void* d_out, int out_size, void* d_ws, size_t ws_size,
                              hipStream_t stream) {
  (void)in_sizes; (void)n_in; (void)out_size; (void)ws_size;

  const float* predict = (const float*)d_in[0];
  const int*   target  = (const int*)d_in[1];
  float*       out     = (float*)d_out;

  // Workspace layout (~16.02 MB):
  //   keys : NPIX * 4 B
  //   nll  : NPIX * 4 B
  //   hist : 256 * 4 B
  //   ctl  : 256 B (padded)
  //   psum : RB * 8 B
  //   pcnt : RB * 4 B
  char* ws = (char*)d_ws;
  unsigned* keys = (unsigned*)ws;
  float*    nll  = (float*)(ws + (size_t)NPIX * 4);
  unsigned* hist = (unsigned*)(ws + (size_t)NPIX * 8);
  Ctl*      ctl  = (Ctl*)(ws + (size_t)NPIX * 8 + 1024);
  double*   psum = (double*)(ws + (size_t)NPIX * 8 + 1024 + 256);
  unsigned* pcnt = (unsigned*)(ws + (size_t)NPIX * 8 + 1024 + 256 + (size_t)RB * 8);

  k_init<<<1, 256, 0, stream>>>(hist, ctl);
  k_prob_nll<<<NPIX / 256, 256, 0, stream>>>(predict, target, keys, nll, ctl);
  k_setup<<<1, 1, 0, stream>>>(ctl);
  for (int shift = 24; shift >= 0; shift -= 8) {
    k_hist<<<NPIX / 4 / 256, 256, 0, stream>>>((const uint4*)keys, ctl, hist, shift);
    k_scan<<<1, 256, 0, stream>>>(ctl, hist, shift);
  }
  k_reduce<<<RB, 256, 0, stream>>>((const uint4*)keys, (const float4*)nll, ctl, psum, pcnt);
  k_final<<<1, 256, 0, stream>>>(psum, pcnt, out);
}